// SubmanifoldSparseConv_36833639530667
// MI455X (gfx1250) — compile-verified
//
#include <hip/hip_runtime.h>
#include <hip/hip_bf16.h>

typedef __attribute__((ext_vector_type(2))) float v2f;
typedef __attribute__((ext_vector_type(8))) float v8f;

#define GRID_G 128
#define CIN 32
#define COUT 32
#define KVOL 27

// ---------------- hash table build ----------------

__global__ void ssc_init_table(int* __restrict__ table, int n) {
    int i = blockIdx.x * blockDim.x + threadIdx.x;
    if (i < n) table[i] = -1;
}

__global__ void ssc_build_table(const int* __restrict__ pos,
                                int* __restrict__ table, int N) {
    int i = blockIdx.x * blockDim.x + threadIdx.x;
    if (i < N) {
        int h = (pos[3 * i] * GRID_G + pos[3 * i + 1]) * GRID_G + pos[3 * i + 2];
        table[h] = i;  // duplicates: arbitrary winner, same as reference scatter-set
    }
}

// ---------------- WMMA sparse conv ----------------
// One wave32 per 16-voxel tile; 8 waves (128 voxels) per block.
// All 27 weight matrices staged in LDS (27*32*32*4 = 110592 B).

__global__ __launch_bounds__(256)
void ssc_conv_wmma(const float* __restrict__ features,
                   const int*   __restrict__ pos,
                   const float* __restrict__ weight,
                   const int*   __restrict__ table,
                   float*       __restrict__ out,
                   int N) {
    extern __shared__ float smem[];  // [KVOL][CIN][COUT]

    // cooperative weight stage: 27648 floats = 6912 float4
    {
        const float4* w4 = (const float4*)weight;
        float4*       s4 = (float4*)smem;
        for (int j = threadIdx.x; j < (KVOL * CIN * COUT) / 4; j += blockDim.x)
            s4[j] = w4[j];
    }
    __syncthreads();

    const int lane = threadIdx.x & 31;
    const int wave = threadIdx.x >> 5;
    const int row  = lane & 15;   // voxel row within tile / matrix M-half
    const int hi   = lane >> 4;   // lane half -> K/M offset per ISA layout
    const int col0 = lane & 15;   // N column for B/C/D fragments

    const int tileBase = blockIdx.x * 128 + wave * 16;
    const int voxel    = tileBase + row;
    const bool rowValid = voxel < N;

    // my voxel position (push invalid rows far out of bounds -> idx = -1)
    int px = -1000, py = -1000, pz = -1000;
    if (rowValid) {
        px = pos[3 * voxel];
        py = pos[3 * voxel + 1];
        pz = pos[3 * voxel + 2];
    }

    v8f c0 = {};  // Cout 0..15
    v8f c1 = {};  // Cout 16..31

    for (int k = 0; k < KVOL; ++k) {
        const int dx = k / 9 - 1;
        const int dy = (k / 3) % 3 - 1;
        const int dz = k % 3 - 1;

        const int nx = px + dx, ny = py + dy, nz = pz + dz;
        const bool inside = ((unsigned)nx < GRID_G) & ((unsigned)ny < GRID_G) &
                            ((unsigned)nz < GRID_G);
        int idx = -1;
        if (inside) idx = table[(nx * GRID_G + ny) * GRID_G + nz];

        const float* fp = features + (size_t)(idx < 0 ? 0 : idx) * CIN;
        const float* wk = smem + k * (CIN * COUT);

        #pragma unroll
        for (int c = 0; c < 8; ++c) {
            const int ch = 4 * c + 2 * hi;  // K pair {ch, ch+1} held by this lane
            // A fragment: gathered neighbor features (zero if inactive)
            v2f a;
            if (idx >= 0) {
                a = *(const v2f*)(fp + ch);
            } else {
                a.x = 0.0f; a.y = 0.0f;
            }
            // B fragments from LDS: W[ch][col], W[ch+1][col] for both N-tiles
            v2f b0, b1;
            b0.x = wk[ch * COUT + col0];
            b0.y = wk[(ch + 1) * COUT + col0];
            b1.x = wk[ch * COUT + col0 + 16];
            b1.y = wk[(ch + 1) * COUT + col0 + 16];

            c0 = __builtin_amdgcn_wmma_f32_16x16x4_f32(
                false, a, false, b0, (short)0, c0, false, false);
            c1 = __builtin_amdgcn_wmma_f32_16x16x4_f32(
                false, a, false, b1, (short)0, c1, false, false);
        }
    }

    // D layout: lane (vgpr r) holds D[M = r + 8*hi][N = col0]
    #pragma unroll
    for (int r = 0; r < 8; ++r) {
        const int vox = tileBase + r + 8 * hi;
        if (vox < N) {
            out[(size_t)vox * COUT + col0]      = c0[r];
            out[(size_t)vox * COUT + col0 + 16] = c1[r];
        }
    }
}

// ---------------- launcher ----------------

extern "C" void kernel_launch(void* const* d_in, const int* in_sizes, int n_in,
                              void* d_out, int out_size, void* d_ws, size_t ws_size,
                              hipStream_t stream) {
    const float* features = (const float*)d_in[0];
    const int*   pos      = (const int*)d_in[1];
    const float* weight   = (const float*)d_in[2];
    float*       out      = (float*)d_out;

    const int N = in_sizes[1] / 3;
    int* table = (int*)d_ws;  // 128^3 ints = 8 MB scratch
    const int T = GRID_G * GRID_G * GRID_G;

    ssc_init_table<<<(T + 255) / 256, 256, 0, stream>>>(table, T);
    ssc_build_table<<<(N + 255) / 256, 256, 0, stream>>>(pos, table, N);

    const int blocks = (N + 127) / 128;
    const size_t lds_bytes = (size_t)KVOL * CIN * COUT * sizeof(float);  // 110592
    ssc_conv_wmma<<<blocks, 256, lds_bytes, stream>>>(features, pos, weight,
                                                      table, out, N);
}